// VectorQuantizer_32100585570378
// MI455X (gfx1250) — compile-verified
//
#include <hip/hip_runtime.h>

// Problem constants (fixed by the reference).
#define KEMB   1024
#define DDIM   64
#define NROWS  65536            // 64*32*32
#define BETAC  0.25f
#define ESTRIDE 68              // padded LDS row stride (floats): +4 banks per row
#define WAVES_PER_BLOCK 8
#define ROWS_PER_WAVE 16
#define ROWS_PER_BLOCK (WAVES_PER_BLOCK * ROWS_PER_WAVE)   // 128
#define NBLOCKS (NROWS / ROWS_PER_BLOCK)                    // 512

typedef __attribute__((ext_vector_type(2))) float v2f;
typedef __attribute__((ext_vector_type(4))) float v4f;
typedef __attribute__((ext_vector_type(8))) float v8f;

#if defined(__has_builtin)
#if __has_builtin(__builtin_amdgcn_global_load_async_to_lds_b128)
#define HAVE_ASYNC_LDS 1
#endif
#endif

#ifdef HAVE_ASYNC_LDS
// Builtin parameter types per the compiler diagnostic: pointer to a 16-byte
// int vector, in the global (AS 1) / LDS (AS 3) address spaces, non-const.
typedef int vq_b128 __attribute__((vector_size(4 * sizeof(int))));
typedef vq_b128 __attribute__((address_space(1))) * vq_g128p;
typedef vq_b128 __attribute__((address_space(3))) * vq_l128p;
#endif

__global__ __launch_bounds__(256) void vq_main_kernel(
    const float* __restrict__ X,     // latents, [NROWS][DDIM]
    const float* __restrict__ E,     // codebook, [KEMB][DDIM]
    float* __restrict__ out,         // quantized_st, [NROWS][DDIM]
    float* __restrict__ partials)    // [NBLOCKS*WAVES_PER_BLOCK] per-wave loss partials
{
    extern __shared__ float smem[];
    float* smE    = smem;                      // [KEMB][ESTRIDE]
    float* smNorm = smem + KEMB * ESTRIDE;     // [KEMB]

    const int tid = threadIdx.x;

    // ---- Stage codebook into LDS (padded stride) ----
#ifdef HAVE_ASYNC_LDS
    for (int i = tid; i < KEMB * (DDIM / 4); i += 256) {
        int row = i >> 4;
        int c4  = (i & 15) * 4;
        const float* g = E + row * DDIM + c4;
        float*       l = smE + row * ESTRIDE + c4;
        // gfx1250 async copy: GLOBAL -> LDS, 16B per lane, tracked by ASYNCcnt
        __builtin_amdgcn_global_load_async_to_lds_b128(
            (vq_g128p)g, (vq_l128p)l, /*imm offset*/0, /*cpol*/0);
    }
#if __has_builtin(__builtin_amdgcn_s_wait_asynccnt)
    __builtin_amdgcn_s_wait_asynccnt(0);
#else
    asm volatile("s_wait_asynccnt 0x0" ::: "memory");
#endif
#else
    for (int i = tid; i < KEMB * (DDIM / 4); i += 256) {
        int row = i >> 4;
        int c4  = (i & 15) * 4;
        v4f v = *(const v4f*)(E + row * DDIM + c4);
        *(v4f*)(smE + row * ESTRIDE + c4) = v;
    }
#endif
    __syncthreads();

    // ---- Per-embedding squared norms ||e||^2 ----
    for (int r = tid; r < KEMB; r += 256) {
        float s = 0.0f;
        #pragma unroll
        for (int c = 0; c < DDIM; ++c) {
            float e = smE[r * ESTRIDE + c];
            s += e * e;
        }
        smNorm[r] = s;
    }
    __syncthreads();

    const int wave = tid >> 5;
    const int lane = tid & 31;
    const int half = lane >> 4;       // 0: lanes 0-15, 1: lanes 16-31
    const int ln   = lane & 15;
    const int k0   = half * 2;        // K-pair offset for A/B striping

    const int rowBase = blockIdx.x * ROWS_PER_BLOCK + wave * ROWS_PER_WAVE;

    // ---- Load A tile (16 rows x 64) into registers in WMMA 16x4 f32 layout ----
    v2f a[16];
    {
        const float* xrow = X + (size_t)(rowBase + ln) * DDIM;
        #pragma unroll
        for (int dt = 0; dt < 16; ++dt)
            a[dt] = *(const v2f*)(xrow + dt * 4 + k0);
    }

    float minD[8];
    int   minI[8];
    #pragma unroll
    for (int j = 0; j < 8; ++j) { minD[j] = 3.4e38f; minI[j] = 0; }

    // ---- Sweep K=1024 embeddings in 16-wide tiles ----
    for (int kt = 0; kt < KEMB / 16; ++kt) {
        // Two independent accumulator chains (K 0..31 and K 32..63) so the
        // matrix pipe can overlap the otherwise serial C-accumulation.
        v8f c0 = {};
        v8f c1 = {};
        const float* bbase = smE + (size_t)(kt * 16 + ln) * ESTRIDE + k0;
        #pragma unroll
        for (int dt = 0; dt < 8; ++dt) {
            v2f b0 = *(const v2f*)(bbase + dt * 4);
            v2f b1 = *(const v2f*)(bbase + (dt + 8) * 4);
            c0 = __builtin_amdgcn_wmma_f32_16x16x4_f32(
                     false, a[dt],     false, b0, (short)0, c0, false, false);
            c1 = __builtin_amdgcn_wmma_f32_16x16x4_f32(
                     false, a[dt + 8], false, b1, (short)0, c1, false, false);
        }
        const float en = smNorm[kt * 16 + ln];
        const int   ei = kt * 16 + ln;
        #pragma unroll
        for (int j = 0; j < 8; ++j) {
            float dot = c0[j] + c1[j];
            float d = en - 2.0f * dot;        // ||e||^2 - 2 x.e  (||x||^2 constant/row)
            if (d < minD[j]) { minD[j] = d; minI[j] = ei; }
        }
    }

    // ---- Cross-lane argmin within each 16-lane group (tie -> smaller index) ----
    #pragma unroll
    for (int j = 0; j < 8; ++j) {
        float d = minD[j];
        int   i = minI[j];
        #pragma unroll
        for (int off = 1; off < 16; off <<= 1) {
            float od = __shfl_xor(d, off, 32);   // masks < 16 stay inside each 16-group
            int   oi = __shfl_xor(i, off, 32);
            if (od < d || (od == d && oi < i)) { d = od; i = oi; }
        }
        minD[j] = d;
        minI[j] = i;
    }

    // ---- Gather emb[idx], straight-through output, loss partial ----
    float lsum = 0.0f;
    #pragma unroll
    for (int j = 0; j < 8; ++j) {
        const int row = rowBase + j + 8 * half;
        const int bi  = minI[j];                 // uniform across the 16-lane group
        v4f q = *(const v4f*)(smE + (size_t)bi * ESTRIDE + ln * 4);
        v4f l = *(const v4f*)(X + (size_t)row * DDIM + ln * 4);
        v4f dq = q - l;
        v4f o  = l + dq;                         // latents + stopgrad(q - latents)
        *(v4f*)(out + (size_t)row * DDIM + ln * 4) = o;
        lsum += dq.x * dq.x + dq.y * dq.y + dq.z * dq.z + dq.w * dq.w;
    }

    // Deterministic wave-level tree reduction of the loss partial.
    #pragma unroll
    for (int off = 1; off < 32; off <<= 1)
        lsum += __shfl_xor(lsum, off, 32);
    if (lane == 0)
        partials[blockIdx.x * WAVES_PER_BLOCK + wave] = lsum;
}

__global__ void vq_finish_kernel(const float* __restrict__ partials,
                                 float* __restrict__ out)
{
    if (threadIdx.x == 0 && blockIdx.x == 0) {
        float s = 0.0f;
        for (int i = 0; i < NBLOCKS * WAVES_PER_BLOCK; ++i)   // fixed order: deterministic
            s += partials[i];
        float m = s / (float)((long long)NROWS * DDIM);
        out[(size_t)NROWS * DDIM] = m * BETAC + m;            // beta*commitment + embedding
    }
}

extern "C" void kernel_launch(void* const* d_in, const int* in_sizes, int n_in,
                              void* d_out, int out_size, void* d_ws, size_t ws_size,
                              hipStream_t stream) {
    (void)in_sizes; (void)n_in; (void)out_size; (void)ws_size;
    const float* X = (const float*)d_in[0];   // latents  [64,32,32,64] f32
    const float* E = (const float*)d_in[1];   // emb_weight [1024,64] f32
    float* out      = (float*)d_out;          // [NROWS*DDIM] quantized_st, then 1 loss float
    float* partials = (float*)d_ws;           // 4096 floats of scratch

    const size_t smem_bytes = (size_t)(KEMB * ESTRIDE + KEMB) * sizeof(float); // ~276 KB LDS
    vq_main_kernel<<<NBLOCKS, 256, smem_bytes, stream>>>(X, E, out, partials);
    vq_finish_kernel<<<1, 32, 0, stream>>>(partials, out);
}